// SymmetryBreakingGNN_40862318854392
// MI455X (gfx1250) — compile-verified
//
#include <hip/hip_runtime.h>
#include <stdint.h>

#define N_NODES 50000
#define N_EDGES 640000
#define CH 128

typedef float v2f __attribute__((ext_vector_type(2)));
typedef float v8f __attribute__((ext_vector_type(8)));

// ---------------------------------------------------------------------------
// Hardware FP32 global atomic add, no return (STOREcnt-tracked).
// Guaranteed global_atomic_add_f32 — dst working set (25.6 MB) lives in L2.
// ---------------------------------------------------------------------------
__device__ __forceinline__ void atomAddF32(float* p, float v) {
  asm volatile("global_atomic_add_f32 %0, %1, off" : : "v"(p), "v"(v)
               : "memory");
}

// ---------------------------------------------------------------------------
// Async DMA: 16 bytes global -> LDS, tracked by ASYNCcnt (bypasses VGPRs).
// ---------------------------------------------------------------------------
__device__ __forceinline__ void asyncLoadLds16(uint32_t lds_byte_off,
                                               const void* gaddr) {
  asm volatile("global_load_async_to_lds_b128 %0, %1, off"
               : : "v"(lds_byte_off), "v"(gaddr) : "memory");
}

__device__ __forceinline__ void waitAsyncCnt0() {
  asm volatile("s_wait_asynccnt 0x0" : : : "memory");
}

// ---------------------------------------------------------------------------
// H[N,128] = act(A[N,128]) @ W[128,128]  via V_WMMA_F32_16X16X4_F32.
// Block = 256 threads = 8 waves. Block b owns rows [16b,16b+16); wave w owns
// columns [16w,16w+16). A stripe DMA'd into LDS (stride 132, conflict-free),
// W (64 KB) read through WGP$/L2.
// ---------------------------------------------------------------------------
template <bool RELU>
__global__ __launch_bounds__(256) void gemm_wmma_f32(
    const float* __restrict__ A, const float* __restrict__ W,
    float* __restrict__ H) {
  __shared__ float sA[16 * 132];

  const int tid  = threadIdx.x;
  const int row0 = blockIdx.x * 16;

  // Stage the 16x128 A stripe via async-to-LDS DMA: 512 float4 chunks,
  // 2 per thread. Chunks never cross the padded row boundary.
#pragma unroll
  for (int i = tid; i < 512; i += 256) {
    const int r  = i >> 5;          // row 0..15
    const int c4 = (i & 31) << 2;   // col 0,4,...,124
    const uint32_t lo = (uint32_t)(uintptr_t)(&sA[r * 132 + c4]);
    asyncLoadLds16(lo, A + (row0 + r) * CH + c4);
  }
  waitAsyncCnt0();
  __syncthreads();

  const int lane = tid & 31;
  const int wave = tid >> 5;
  const int n    = lane & 15;  // N index (and A's M index for this lane)
  const int hi   = lane >> 4;  // lane half selects K pair {0,1} vs {2,3}
  const int col0 = wave * 16;

  v8f acc = {};
#pragma unroll
  for (int k0 = 0; k0 < CH; k0 += 4) {
    const int ka = k0 + hi * 2;
    v2f a, b;
    a.x = sA[n * 132 + ka];      // A[m=n][ka]
    a.y = sA[n * 132 + ka + 1];  // A[m=n][ka+1]
    if (RELU) {
      a.x = fmaxf(a.x, 0.0f);
      a.y = fmaxf(a.y, 0.0f);
    }
    b.x = W[ka * CH + col0 + n];        // W[ka][col]
    b.y = W[(ka + 1) * CH + col0 + n];  // W[ka+1][col]
    acc = __builtin_amdgcn_wmma_f32_16x16x4_f32(
        /*neg_a=*/false, a, /*neg_b=*/false, b,
        /*c_mod=*/(short)0, acc, /*reuse_a=*/false, /*reuse_b=*/false);
  }

  // D layout: VGPR r -> row (hi*8 + r), col n.
  float* out = H + (row0 + hi * 8) * CH + col0 + n;
#pragma unroll
  for (int r = 0; r < 8; ++r) out[r * CH] = acc[r];
}

// ---------------------------------------------------------------------------
// X[i, c] = bias[c]  (segment_sum accumulator init; out = agg + b).
// ---------------------------------------------------------------------------
__global__ __launch_bounds__(256) void init_bias(float* __restrict__ X,
                                                 const float* __restrict__ b) {
  const int i = blockIdx.x * 256 + threadIdx.x;
  X[i] = b[i & (CH - 1)];
}

// ---------------------------------------------------------------------------
// Wave-per-edge scatter-add: X[dst] += H[src]. Each lane moves one float4
// (128 ch / 32 lanes) with 4 hardware f32 atomics resolved in L2.
// ---------------------------------------------------------------------------
__global__ __launch_bounds__(256) void scatter_add(
    const float* __restrict__ H, const int* __restrict__ edge,
    float* __restrict__ X) {
  const int lane = threadIdx.x & 31;
  const int wave = threadIdx.x >> 5;
  const int e    = blockIdx.x * 8 + wave;

  const int src = edge[e];            // edge_index[0][e]
  const int dst = edge[N_EDGES + e];  // edge_index[1][e]

  const float4 v = ((const float4*)(H + src * CH))[lane];
  float* p = X + dst * CH + lane * 4;
  atomAddF32(p + 0, v.x);
  atomAddF32(p + 1, v.y);
  atomAddF32(p + 2, v.z);
  atomAddF32(p + 3, v.w);
}

// ---------------------------------------------------------------------------
extern "C" void kernel_launch(void* const* d_in, const int* in_sizes, int n_in,
                              void* d_out, int out_size, void* d_ws,
                              size_t ws_size, hipStream_t stream) {
  const float* v0   = (const float*)d_in[0];
  const int*   edge = (const int*)d_in[1];
  const float* W1   = (const float*)d_in[2];
  const float* b1   = (const float*)d_in[3];
  const float* W2   = (const float*)d_in[4];
  const float* b2   = (const float*)d_in[5];
  float*       out  = (float*)d_out;

  const size_t feat = (size_t)N_NODES * CH;
  float* H1 = (float*)d_ws;  // [N,128] dense transform, layer 1
  float* X1 = H1 + feat;     // [N,128] aggregated + bias, layer 1
  float* H2 = X1 + feat;     // [N,128] dense transform, layer 2

  const int gemmBlocks = N_NODES / 16;          // 3125
  const int initBlocks = (N_NODES * CH) / 256;  // 25000
  const int edgeBlocks = N_EDGES / 8;           // 80000

  // Layer 1: H1 = v0 @ W1 ; X1 = segment_sum(H1[src] -> dst) + b1
  gemm_wmma_f32<false><<<gemmBlocks, 256, 0, stream>>>(v0, W1, H1);
  init_bias<<<initBlocks, 256, 0, stream>>>(X1, b1);
  scatter_add<<<edgeBlocks, 256, 0, stream>>>(H1, edge, X1);

  // Layer 2: H2 = relu(X1) @ W2 ; out = segment_sum(H2[src] -> dst) + b2
  gemm_wmma_f32<true><<<gemmBlocks, 256, 0, stream>>>(X1, W2, H2);
  init_bias<<<initBlocks, 256, 0, stream>>>(out, b2);
  scatter_add<<<edgeBlocks, 256, 0, stream>>>(H2, edge, out);
}